// FCCaps_5480378270383
// MI455X (gfx1250) — compile-verified
//
#include <hip/hip_runtime.h>
#include <hip/hip_bf16.h>
#include <math.h>
#include <stdint.h>

// FCCaps (capsule routing) for MI455X / gfx1250, fp32 WMMA path.
//   B=1024, iN=J=128, iD=16, N=32, D=32
// Kernels (all on `stream`):
//   caps_repackW: one-time pack of W into per-lane WMMA B-fragment order so
//                 every fragment is a single coalesced global_load_b64
//   caps_usum   : partial usum[q][b][n][d] over j-chunks (v_wmma_f32_16x16x4_f32)
//   caps_route  : recompute u per j, dot with usum, softmax over n in LDS,
//                 accumulate partial s[q][b][n][d]
//   caps_squash : sum partials, squash, write out
// x tiles are DMA'd into LDS by the Tensor Data Mover (tensor_load_to_lds)
// with LDS padding (2 dwords per 256) so A-fragment ds_load_b64 reads are
// bank-conflict-free.
// Deterministic: no float atomics; per-chunk partial buffers in d_ws.
// d_ws requirement: 8 MB (Wt) + 2 * JS * B*N*D * 4 B (32 MB) = 40 MB.

typedef float v8f  __attribute__((ext_vector_type(8)));
typedef float v2f  __attribute__((ext_vector_type(2)));
typedef unsigned int u32x4 __attribute__((ext_vector_type(4)));
typedef int   i32x8 __attribute__((ext_vector_type(8)));
typedef int   i32x4 __attribute__((ext_vector_type(4)));

#define B_   1024
#define J_   128
#define ID_  16
#define N_   32
#define D_   32
#define BT_  16                 // batch tile (WMMA M)
#define JS_  4                  // j-split for machine fill
#define JC_  (J_/JS_)           // 32 j per block
#define BND_ (B_*N_*D_)         // 1048576 elements per partial buffer
#define WT_PAIRS_ (N_*J_*256)   // 1048576 float2 pairs (8 MB)
#define XLDS_ (JC_*ID_*BT_ + 2*(JC_*ID_*BT_/256))  // 8192 + 64 padded dwords

// ---- TDM preload of one x tile (16 rows x 512 floats, row stride 2048) ----
// LDS layout: addr_dw(flat) = flat + 2*(flat>>8)  (TDM pad: 2 dwords / 256)
static __device__ inline void preload_x(const float* __restrict__ gsrc,
                                        float* ldsX) {
#if __has_builtin(__builtin_amdgcn_tensor_load_to_lds) && \
    __has_builtin(__builtin_amdgcn_s_wait_tensorcnt)
  if (threadIdx.x < 32) {           // one wave issues the DMA (EXEC ignored)
    const unsigned long long ga = (unsigned long long)(uintptr_t)gsrc;
    const unsigned lds = (unsigned)(uintptr_t)ldsX;  // flat low 32 = LDS offset
    u32x4 g0;
    g0[0] = 1u;                                            // count=1, user mode
    g0[1] = lds;                                           // lds_addr (bytes)
    g0[2] = (unsigned)(ga & 0xFFFFFFFFu);                  // global_addr lo
    g0[3] = (unsigned)((ga >> 32) & 0x01FFFFFFu) | (2u << 30);  // hi | type=2
    i32x8 g1;
    g1[0] = (2 << 16)        // data_size = 4 bytes
          | (1 << 20)        // pad_enable
          | (7 << 22)        // pad_interval code 7 = 256 dwords
          | (1 << 25);       // pad_amount  code 1 = 2 dwords
    g1[1] = (int)(512u << 16);   // tensor_dim0 = 512 elements (bits 79:48 lo)
    g1[2] = (int)(16u << 16);    // tensor_dim0 hi=0 | tensor_dim1 = 16
    g1[3] = (int)(512u << 16);   // tensor_dim1 hi=0 | tile_dim0 = 512
    g1[4] = 16;                  // tile_dim1 = 16, tile_dim2 = 0
    g1[5] = 2048;                // tensor_dim0_stride = 2048 elements (lo)
    g1[6] = 0;                   // stride hi / dim1_stride lo
    g1[7] = 0;
    i32x4 g2 = {0, 0, 0, 0}, g3 = {0, 0, 0, 0};
    i32x8 g4 = {0, 0, 0, 0, 0, 0, 0, 0};   // trailing group (6-arg builtin form)
    __builtin_amdgcn_tensor_load_to_lds(g0, g1, g2, g3, g4, 0);
    __builtin_amdgcn_s_wait_tensorcnt(0);
  }
#else
  for (int flat = (int)threadIdx.x; flat < JC_*ID_*BT_; flat += 256) {
    const int b = flat >> 9, r = flat & 511;
    ldsX[flat + 2*(flat >> 8)] = gsrc[(size_t)b*(J_*ID_) + r];
  }
#endif
}

// ---------------------------------------------------------- W repack kernel
// Wt[((n*J + j)*8 + h*4 + k)*32 + lane] = { W[n,j,i0,d], W[n,j,i0+1,d] }
//   i0 = 4k + 2*(lane>>4), d = (lane&15) + 16h  -> one b64 per fragment lane.
__global__ __launch_bounds__(256) void caps_repackW(
    const float* __restrict__ W, float2* __restrict__ Wt) {
  const int p = blockIdx.x*256 + threadIdx.x;
  if (p >= WT_PAIRS_) return;
  const int lane = p & 31;
  const int kk   = (p >> 5) & 7;      // h*4 + k
  const int nj   = p >> 8;
  const int h = kk >> 2, k = kk & 3;
  const int i0 = 4*k + 2*(lane >> 4);
  const int d  = (lane & 15) + 16*h;
  const float* src = W + (size_t)nj * (ID_*D_);
  Wt[p] = make_float2(src[i0*D_ + d], src[(i0 + 1)*D_ + d]);
}

// ---------------------------------------------------------------- phase A
__global__ __launch_bounds__(256) void caps_usum(
    const float* __restrict__ x, const float2* __restrict__ Wt,
    float* __restrict__ usumPart) {
  __shared__ __align__(16) float ldsX[XLDS_];
  const int b0 = blockIdx.x * BT_;
  const int j0 = blockIdx.y * JC_;

  preload_x(x + ((size_t)b0*J_ + j0)*ID_, ldsX);
  __syncthreads();

  const int t = threadIdx.x;
  const int wave = t >> 5, lane = t & 31;
  const int m = lane & 15, hi = lane >> 4;

  v8f acc[4][2] = {};  // usum accumulators: 4 n-values x 2 D-halves

  for (int jj = 0; jj < JC_; ++jj) {
    const int j = j0 + jj;
    v2f a[4];
#pragma unroll
    for (int k = 0; k < 4; ++k) {     // conflict-free ds_load_b64 per lane
      const int flat = m*512 + jj*ID_ + 4*k + 2*hi;
      a[k] = *(const v2f*)&ldsX[flat + 2*(flat >> 8)];
    }
#pragma unroll
    for (int nn = 0; nn < 4; ++nn) {
      const int n = wave*4 + nn;
      const v2f* Wb = (const v2f*)Wt + ((size_t)(n*J_ + j))*256 + lane;
      __builtin_prefetch(Wb + 256, 0, 3);     // next j's fragments -> near cache
#pragma unroll
      for (int h = 0; h < 2; ++h)
#pragma unroll
        for (int k = 0; k < 4; ++k)
          acc[nn][h] = __builtin_amdgcn_wmma_f32_16x16x4_f32(
              false, a[k], false, Wb[(h*4 + k)*32], (short)0,
              acc[nn][h], false, false);
    }
  }

  // C/D layout: lane half hi, vgpr v -> batch row v+8*hi; lane&15 + 16*h -> d
  float* dst = usumPart + (size_t)blockIdx.y * BND_;
#pragma unroll
  for (int nn = 0; nn < 4; ++nn) {
    const int n = wave*4 + nn;
#pragma unroll
    for (int h = 0; h < 2; ++h)
#pragma unroll
      for (int v = 0; v < 8; ++v)
        dst[(size_t)(b0 + v + 8*hi)*(N_*D_) + n*D_ + (m + 16*h)] = acc[nn][h][v];
  }
}

// ---------------------------------------------------------------- phase B
__global__ __launch_bounds__(256) void caps_route(
    const float* __restrict__ x, const float2* __restrict__ Wt,
    const float* __restrict__ bias, const float* __restrict__ usumPart,
    float* __restrict__ sPart) {
  __shared__ __align__(16) float ldsX[XLDS_];
  __shared__ float ldsBias[N_*JC_];     // [n][jj]
  __shared__ float ldsC[N_*BT_];        // routing logits/weights [n][b]
  const int b0 = blockIdx.x * BT_;
  const int j0 = blockIdx.y * JC_;
  const int t  = threadIdx.x;

  preload_x(x + ((size_t)b0*J_ + j0)*ID_, ldsX);
  for (int idx = t; idx < N_*JC_; idx += 256) {
    const int n = idx / JC_, jj = idx % JC_;
    ldsBias[idx] = bias[n*J_ + (j0 + jj)];
  }
  __syncthreads();

  const int wave = t >> 5, lane = t & 31;
  const int m = lane & 15, hi = lane >> 4;

  // gather usum (deterministic sum over JS_ partials), j-independent
  v8f us[4][2];
#pragma unroll
  for (int nn = 0; nn < 4; ++nn)
#pragma unroll
    for (int h = 0; h < 2; ++h)
#pragma unroll
      for (int v = 0; v < 8; ++v) {
        const size_t gi = (size_t)(b0 + v + 8*hi)*(N_*D_) + (wave*4 + nn)*D_ + (m + 16*h);
        float s0 = 0.f;
#pragma unroll
        for (int q = 0; q < JS_; ++q) s0 += usumPart[(size_t)q*BND_ + gi];
        us[nn][h][v] = s0;
      }

  v8f sacc[4][2] = {};
  const float rsD = 0.17677669529663689f;   // 1/sqrt(D=32)

  for (int jj = 0; jj < JC_; ++jj) {
    const int j = j0 + jj;
    v2f a[4];
#pragma unroll
    for (int k = 0; k < 4; ++k) {
      const int flat = m*512 + jj*ID_ + 4*k + 2*hi;
      a[k] = *(const v2f*)&ldsX[flat + 2*(flat >> 8)];
    }
    v8f u[4][2] = {};
#pragma unroll
    for (int nn = 0; nn < 4; ++nn) {
      const int n = wave*4 + nn;
      const v2f* Wb = (const v2f*)Wt + ((size_t)(n*J_ + j))*256 + lane;
      __builtin_prefetch(Wb + 256, 0, 3);
#pragma unroll
      for (int h = 0; h < 2; ++h)
#pragma unroll
        for (int k = 0; k < 4; ++k)
          u[nn][h] = __builtin_amdgcn_wmma_f32_16x16x4_f32(
              false, a[k], false, Wb[(h*4 + k)*32], (short)0,
              u[nn][h], false, false);
    }

    // c[b,n,j] = (u . usum)/sqrt(D): per-lane partial + 16-lane xor reduce
#pragma unroll
    for (int nn = 0; nn < 4; ++nn) {
      float p[8];
#pragma unroll
      for (int v = 0; v < 8; ++v)
        p[v] = u[nn][0][v]*us[nn][0][v] + u[nn][1][v]*us[nn][1][v];
#pragma unroll
      for (int off = 1; off < 16; off <<= 1)
#pragma unroll
        for (int v = 0; v < 8; ++v) p[v] += __shfl_xor(p[v], off, 32);
      if (m == 0) {
#pragma unroll
        for (int v = 0; v < 8; ++v)
          ldsC[(wave*4 + nn)*BT_ + (v + 8*hi)] = p[v] * rsD;
      }
    }
    __syncthreads();

    // softmax over n per batch element, then + bias (one thread per b)
    if (t < BT_) {
      float mx = -3.0e38f;
      for (int n2 = 0; n2 < N_; ++n2) mx = fmaxf(mx, ldsC[n2*BT_ + t]);
      float Z = 0.f;
      for (int n2 = 0; n2 < N_; ++n2) {
        const float e = expf(ldsC[n2*BT_ + t] - mx);
        ldsC[n2*BT_ + t] = e;
        Z += e;
      }
      const float iZ = 1.f / Z;
      for (int n2 = 0; n2 < N_; ++n2)
        ldsC[n2*BT_ + t] = ldsC[n2*BT_ + t]*iZ + ldsBias[n2*JC_ + jj];
    }
    __syncthreads();

    // s += c~ * u
#pragma unroll
    for (int nn = 0; nn < 4; ++nn) {
      float ct[8];
#pragma unroll
      for (int v = 0; v < 8; ++v)
        ct[v] = ldsC[(wave*4 + nn)*BT_ + (v + 8*hi)];
#pragma unroll
      for (int h = 0; h < 2; ++h)
#pragma unroll
        for (int v = 0; v < 8; ++v)
          sacc[nn][h][v] += ct[v] * u[nn][h][v];
    }
    __syncthreads();   // ldsC reused next j
  }

  float* dst = sPart + (size_t)blockIdx.y * BND_;
#pragma unroll
  for (int nn = 0; nn < 4; ++nn) {
    const int n = wave*4 + nn;
#pragma unroll
    for (int h = 0; h < 2; ++h)
#pragma unroll
      for (int v = 0; v < 8; ++v)
        dst[(size_t)(b0 + v + 8*hi)*(N_*D_) + n*D_ + (m + 16*h)] = sacc[nn][h][v];
  }
}

// ---------------------------------------------------------------- phase C
__global__ __launch_bounds__(256) void caps_squash(
    const float* __restrict__ sPart, float* __restrict__ out) {
  const int idx = blockIdx.x * 256 + threadIdx.x;   // (b,n)
  if (idx >= B_*N_) return;
  float sv[D_];
#pragma unroll
  for (int d = 0; d < D_; ++d) sv[d] = 0.f;
  for (int q = 0; q < JS_; ++q) {
    const float4* p = (const float4*)(sPart + (size_t)q*BND_ + (size_t)idx*D_);
#pragma unroll
    for (int dq = 0; dq < D_/4; ++dq) {
      const float4 v = p[dq];
      sv[dq*4 + 0] += v.x; sv[dq*4 + 1] += v.y;
      sv[dq*4 + 2] += v.z; sv[dq*4 + 3] += v.w;
    }
  }
  float nrm2 = 0.f;
#pragma unroll
  for (int d = 0; d < D_; ++d) nrm2 += sv[d]*sv[d];
  const float nrm = sqrtf(nrm2);
  const float f = (1.f - 1.f/(expf(nrm) + 1e-20f)) / (nrm + 1e-20f);
  float4* o = (float4*)(out + (size_t)idx*D_);
#pragma unroll
  for (int dq = 0; dq < D_/4; ++dq)
    o[dq] = make_float4(f*sv[dq*4+0], f*sv[dq*4+1], f*sv[dq*4+2], f*sv[dq*4+3]);
}

extern "C" void kernel_launch(void* const* d_in, const int* in_sizes, int n_in,
                              void* d_out, int out_size, void* d_ws, size_t ws_size,
                              hipStream_t stream) {
  const float* x    = (const float*)d_in[0];   // [1024,128,16]
  const float* W    = (const float*)d_in[1];   // [32,128,16,32]
  const float* bias = (const float*)d_in[2];   // [32,128,1]
  float* out = (float*)d_out;                  // [1024,32,32]

  float2* Wt      = (float2*)d_ws;                      // 8 MB packed W
  float* usumPart = (float*)d_ws + 2*(size_t)WT_PAIRS_; // 16 MB
  float* sPart    = usumPart + (size_t)JS_ * BND_;      // 16 MB

  caps_repackW<<<WT_PAIRS_/256, 256, 0, stream>>>(W, Wt);

  dim3 grid(B_/BT_, JS_);   // 64 batch tiles x 4 j-chunks = 256 WGs
  caps_usum  <<<grid, 256, 0, stream>>>(x, Wt, usumPart);
  caps_route <<<grid, 256, 0, stream>>>(x, Wt, bias, usumPart, sPart);
  caps_squash<<<(B_*N_ + 255)/256, 256, 0, stream>>>(sPart, out);
}